// GraphAttentionLayer_1726576853711
// MI455X (gfx1250) — compile-verified
//
#include <hip/hip_runtime.h>
#include <hip/hip_bf16.h>

typedef float v2f __attribute__((ext_vector_type(2)));
typedef float v8f __attribute__((ext_vector_type(8)));

#define ALPHA 0.2f
#define N_NODES 8192
#define F_IN 256
#define F_OUT 64
#define NSEG 8
#define SEG_LEN (N_NODES / NSEG)   // 1024 j per segment

// ---------------------------------------------------------------------------
// Kernel 1: h = x @ W  (8192x256 @ 256x64) via V_WMMA_F32_16X16X4_F32.
// One wave computes one 16x16 tile of h. 512 row-tiles * 4 col-tiles = 2048.
// A layout (16x4 f32): lanes 0-15 -> K=0,1 ; lanes 16-31 -> K=2,3 (vgpr = K&1)
// B layout (4x16 f32): vgpr0 -> K=2*kh, vgpr1 -> K=2*kh+1, N = lane&15
// ---------------------------------------------------------------------------
__global__ void k_gemm_xw(const float* __restrict__ x, const float* __restrict__ W,
                          float* __restrict__ h) {
  const int lane = threadIdx.x & 31;
  const int wave = threadIdx.x >> 5;
  const int tile = blockIdx.x * (blockDim.x >> 5) + wave;   // 0..2047
  const int mt = tile >> 2;
  const int ct = tile & 3;
  const int m  = lane & 15;
  const int kh = lane >> 4;
  const int row = mt * 16 + m;
  const int col = ct * 16 + m;

  v8f c = {};
  for (int k = 0; k < F_IN; k += 4) {
    const int ka = k + 2 * kh;
    v2f a = *(const v2f*)(x + (size_t)row * F_IN + ka);
    v2f b;
    b.x = W[(ka + 0) * F_OUT + col];
    b.y = W[(ka + 1) * F_OUT + col];
    c = __builtin_amdgcn_wmma_f32_16x16x4_f32(false, a, false, b, (short)0, c,
                                              false, false);
  }
  const int rb = mt * 16 + 8 * kh;
#pragma unroll
  for (int r = 0; r < 8; ++r)
    h[(size_t)(rb + r) * F_OUT + col] = c[r];
}

// ---------------------------------------------------------------------------
// Kernel 2: s1[i] = h[i,:] . a1 ; s2[i] = h[i,:] . a2
// ---------------------------------------------------------------------------
__global__ void k_scores(const float* __restrict__ h, const float* __restrict__ a1,
                         const float* __restrict__ a2, float* __restrict__ s1,
                         float* __restrict__ s2) {
  const int i = blockIdx.x * blockDim.x + threadIdx.x;
  if (i >= N_NODES) return;
  float acc1 = 0.f, acc2 = 0.f;
  const float* hr = h + (size_t)i * F_OUT;
#pragma unroll 8
  for (int n = 0; n < F_OUT; ++n) {
    const float v = hr[n];
    acc1 += v * a1[n];
    acc2 += v * a2[n];
  }
  s1[i] = acc1;
  s2[i] = acc2;
}

// ---------------------------------------------------------------------------
// Kernel 3: smax = max_j s2[j]  (softmax shift; leakyrelu is monotone so
// leakyrelu(s1_i + smax) upper-bounds every masked row max)
// ---------------------------------------------------------------------------
__global__ void k_smax(const float* __restrict__ s2, float* __restrict__ smax) {
  __shared__ float red[256];
  float m = -3.4e38f;
  for (int i = threadIdx.x; i < N_NODES; i += 256) m = fmaxf(m, s2[i]);
  red[threadIdx.x] = m;
  __syncthreads();
  for (int s = 128; s > 0; s >>= 1) {
    if (threadIdx.x < s) red[threadIdx.x] = fmaxf(red[threadIdx.x], red[threadIdx.x + s]);
    __syncthreads();
  }
  if (threadIdx.x == 0) *smax = red[0];
}

// ---------------------------------------------------------------------------
// Kernel 4: fused masked softmax numerator + partial (P @ h), per j-segment.
// wave = (row-tile rt, segment seg): 512*8 = 4096 waves.
// Per iteration (8 j): one b128 adj load + one b128 s2 load per lane.
// K->j mapping inside each WMMA tile is permuted so lane j's are contiguous:
//   tile0: K0..K3 -> jb+0, jb+1, jb+4, jb+5
//   tile1: K0..K3 -> jb+2, jb+3, jb+6, jb+7
// (lane kh=0 supplies K0/K1, kh=1 supplies K2/K3 -> lane j's = jl..jl+3,
//  jl = jb + 4*kh, for both A and B -> consistent, order-invariant sum.)
// ---------------------------------------------------------------------------
__global__ void k_attn_partial(const int* __restrict__ adj, const float* __restrict__ h,
                               const float* __restrict__ s1, const float* __restrict__ s2,
                               const float* __restrict__ smaxp,
                               float* __restrict__ accP, float* __restrict__ denP) {
  const int lane = threadIdx.x & 31;
  const int wave = threadIdx.x >> 5;
  const int wg   = blockIdx.x * (blockDim.x >> 5) + wave;   // 0..4095
  const int rt   = wg >> 3;                                 // row tile 0..511
  const int seg  = wg & (NSEG - 1);                         // j segment 0..7
  const int m  = lane & 15;
  const int kh = lane >> 4;
  const int row = rt * 16 + m;

  const float smax = *smaxp;
  const float s1v  = s1[row];
  float t0 = s1v + smax;
  const float Mv = t0 > 0.f ? t0 : ALPHA * t0;

  v8f acc0 = {}, acc1 = {}, acc2 = {}, acc3 = {};
  float denom = 0.f;
  const size_t rowBase = (size_t)row * N_NODES;
  const int jBeg = seg * SEG_LEN, jEnd = jBeg + SEG_LEN;

  for (int jb = jBeg; jb < jEnd; jb += 8) {
    const int jl = jb + 4 * kh;                             // lane's 4 contiguous j
    const int4   av = *(const int4*)(adj + rowBase + jl);
    const float4 sv = *(const float4*)(s2 + jl);
    __builtin_prefetch(adj + rowBase + jl + 512, 0, 0);     // 2 KB ahead on adj stream

    float e0 = s1v + sv.x; e0 = e0 > 0.f ? e0 : ALPHA * e0;
    float e1 = s1v + sv.y; e1 = e1 > 0.f ? e1 : ALPHA * e1;
    float e2 = s1v + sv.z; e2 = e2 > 0.f ? e2 : ALPHA * e2;
    float e3 = s1v + sv.w; e3 = e3 > 0.f ? e3 : ALPHA * e3;
    const float p0 = (av.x > 0) ? __expf(e0 - Mv) : 0.f;
    const float p1 = (av.y > 0) ? __expf(e1 - Mv) : 0.f;
    const float p2 = (av.z > 0) ? __expf(e2 - Mv) : 0.f;
    const float p3 = (av.w > 0) ? __expf(e3 - Mv) : 0.f;
    denom += (p0 + p1) + (p2 + p3);

    v2f aT0; aT0.x = p0; aT0.y = p1;                        // A tile 0
    v2f aT1; aT1.x = p2; aT1.y = p3;                        // A tile 1

    const float* hb = h + (size_t)jl * F_OUT + m;           // rows jl..jl+3 of h
    v2f b00, b01, b02, b03, b10, b11, b12, b13;
    b00.x = hb[0];          b00.y = hb[F_OUT];              // tile0, cols 0..15
    b01.x = hb[16];         b01.y = hb[F_OUT + 16];
    b02.x = hb[32];         b02.y = hb[F_OUT + 32];
    b03.x = hb[48];         b03.y = hb[F_OUT + 48];
    b10.x = hb[2 * F_OUT];      b10.y = hb[3 * F_OUT];      // tile1, cols 0..15
    b11.x = hb[2 * F_OUT + 16]; b11.y = hb[3 * F_OUT + 16];
    b12.x = hb[2 * F_OUT + 32]; b12.y = hb[3 * F_OUT + 32];
    b13.x = hb[2 * F_OUT + 48]; b13.y = hb[3 * F_OUT + 48];

    acc0 = __builtin_amdgcn_wmma_f32_16x16x4_f32(false, aT0, false, b00, (short)0, acc0, false, false);
    acc1 = __builtin_amdgcn_wmma_f32_16x16x4_f32(false, aT0, false, b01, (short)0, acc1, false, false);
    acc2 = __builtin_amdgcn_wmma_f32_16x16x4_f32(false, aT0, false, b02, (short)0, acc2, false, false);
    acc3 = __builtin_amdgcn_wmma_f32_16x16x4_f32(false, aT0, false, b03, (short)0, acc3, false, false);
    acc0 = __builtin_amdgcn_wmma_f32_16x16x4_f32(false, aT1, false, b10, (short)0, acc0, false, false);
    acc1 = __builtin_amdgcn_wmma_f32_16x16x4_f32(false, aT1, false, b11, (short)0, acc1, false, false);
    acc2 = __builtin_amdgcn_wmma_f32_16x16x4_f32(false, aT1, false, b12, (short)0, acc2, false, false);
    acc3 = __builtin_amdgcn_wmma_f32_16x16x4_f32(false, aT1, false, b13, (short)0, acc3, false, false);
  }

  // per-row partial denominator (row m lives in lanes m and m+16)
  denom += __shfl_xor(denom, 16);
  if (kh == 0) denP[(size_t)seg * N_NODES + row] = denom;

  // store unnormalized partial numerator; D VGPR r -> row (r + 8*kh)
  float* ap = accP + (size_t)seg * N_NODES * F_OUT;
#pragma unroll
  for (int r = 0; r < 8; ++r) {
    const size_t orow = (size_t)(rt * 16 + r + 8 * kh) * F_OUT;
    ap[orow +  0 + m] = acc0[r];
    ap[orow + 16 + m] = acc1[r];
    ap[orow + 32 + m] = acc2[r];
    ap[orow + 48 + m] = acc3[r];
  }
}

// ---------------------------------------------------------------------------
// Kernel 5: combine segment partials, normalize, ELU.
// ---------------------------------------------------------------------------
__global__ void k_finalize(const float* __restrict__ accP, const float* __restrict__ denP,
                           float* __restrict__ out) {
  const int id = blockIdx.x * blockDim.x + threadIdx.x;
  if (id >= N_NODES * F_OUT) return;
  const int row = id >> 6;
  float num = 0.f, den = 0.f;
#pragma unroll
  for (int s = 0; s < NSEG; ++s) {
    num += accP[(size_t)s * N_NODES * F_OUT + id];
    den += denP[(size_t)s * N_NODES + row];
  }
  const float o = num / den;
  out[id] = o > 0.f ? o : (__expf(o) - 1.f);
}

// ---------------------------------------------------------------------------
extern "C" void kernel_launch(void* const* d_in, const int* in_sizes, int n_in,
                              void* d_out, int out_size, void* d_ws, size_t ws_size,
                              hipStream_t stream) {
  const float* x   = (const float*)d_in[0];   // 8192 x 256
  const int*   adj = (const int*)  d_in[1];   // 8192 x 8192
  const float* W   = (const float*)d_in[2];   // 256 x 64
  const float* a1  = (const float*)d_in[3];   // 64
  const float* a2  = (const float*)d_in[4];   // 64
  float* out = (float*)d_out;                 // 8192 x 64

  // workspace layout (floats): h | s1 | s2 | smax(+pad) | accP | denP  (~19 MB)
  float* f    = (float*)d_ws;
  float* h    = f;                                   // 8192*64
  float* s1   = h + (size_t)N_NODES * F_OUT;         // 8192
  float* s2   = s1 + N_NODES;                        // 8192
  float* smax = s2 + N_NODES;                        // 64 (padded)
  float* accP = smax + 64;                           // NSEG * 8192*64
  float* denP = accP + (size_t)NSEG * N_NODES * F_OUT; // NSEG * 8192

  k_gemm_xw<<<256, 256, 0, stream>>>(x, W, h);
  k_scores<<<(N_NODES + 255) / 256, 256, 0, stream>>>(h, a1, a2, s1, s2);
  k_smax<<<1, 256, 0, stream>>>(s2, smax);
  // 4096 waves (512 row-tiles x 8 segments), 4 waves per block
  k_attn_partial<<<1024, 128, 0, stream>>>(adj, h, s1, s2, smax, accP, denP);
  k_finalize<<<(N_NODES * F_OUT + 255) / 256, 256, 0, stream>>>(accP, denP, out);
}